// Black_oil_87514253623720
// MI455X (gfx1250) — compile-verified
//
#include <hip/hip_runtime.h>
#include <stdint.h>

// ---------------- problem geometry (from reference setup_inputs) ----------------
#define B_  16
#define T_  60
#define NX  128
#define NY  128
#define TX  16            // x-rows per tile
#define TT  4             // time steps pipelined per block (60 % 4 == 0)
#define HROWS (TX + 2)    // +1 halo row each side (replicate-clamped)

// ---------------- folded constants -----------------
// dxf = 1/NY = 1/128 ; fd1 scale = 0.5/dxf = 64 ; fd2 scale = 1/dxf^2 = 16384
#define FD1    64.0f
#define FD2    16384.0f
#define PINIA  600.0f              // u = pressure * PINI_ALT
#define SCALE  (1e-7f / 128.0f)    // dxf * 1e-7

// CDNA5 async global->LDS copy (ASYNCcnt-tracked), 16B per lane.
// Inline asm: portable across ROCm7.2 / amdgpu-toolchain (bypasses builtin arity skew).
__device__ __forceinline__ void async_ld_b128(uint32_t lds_off, uint64_t gaddr) {
  asm volatile("global_load_async_to_lds_b128 %0, %1, off"
               :: "v"(lds_off), "v"(gaddr) : "memory");
}
__device__ __forceinline__ void async_wait0() {
#if __has_builtin(__builtin_amdgcn_s_wait_asynccnt)
  __builtin_amdgcn_s_wait_asynccnt(0);
#else
  asm volatile("s_wait_asynccnt 0" ::: "memory");
#endif
}

__global__ __launch_bounds__(256) void black_oil_kernel(
    const float* __restrict__ pressure, const float* __restrict__ perm,
    const float* __restrict__ Q,        const float* __restrict__ Qw,
    const float* __restrict__ Timev,    const float* __restrict__ Phi,
    const float* __restrict__ Swini,    const float* __restrict__ wsat,
    float* __restrict__ out_p, float* __restrict__ out_s)
{
  __shared__ float s_perm[HROWS * NY];
  __shared__ float s_p[2][HROWS * NY];   // ping-pong pressure tiles

  const int tid  = threadIdx.x;
  const int lane = tid & 31;
  const int wid  = tid >> 5;             // 0..7 (8 wave32 per block)
  const int x0   = blockIdx.x * TX;
  const int t0   = blockIdx.y * TT;
  const int b    = blockIdx.z;

  // LDS byte offsets of the shared buffers (one generic->offset conversion each)
  const uint32_t l_perm = (uint32_t)(uintptr_t)&s_perm[0];
  const uint32_t l_p0   = (uint32_t)(uintptr_t)&s_p[0][0];
  const uint32_t l_p1   = (uint32_t)(uintptr_t)&s_p[1][0];

  // siniuse = Swini[0,0,0,0]  (scalar; prior_sat at t==0)
  const float siniuse = Swini[0];
  const float S0  = (siniuse - 0.1f) * 1.25f;        // /(1-SWI-SWR)=*1.25
  const float c0  = (S0 * S0 + (1.0f - S0) * (1.0f - S0) * (1.0f / 2.75f)) * 500.0f; // a1[t=0]  = c0 *perm
  const float cw0 = (S0 * S0) * 500.0f;                                              // a1w[t=0] = cw0*perm

  // ---- async stage: perm tile + pressure tile(t0) -> LDS (halo rows clamped) ----
  const uint32_t lrow = (uint32_t)(lane << 4);       // 16B chunk per lane within a row
  for (int r = wid; r < HROWS; r += 8) {
    int gx = x0 - 1 + r; gx = gx < 0 ? 0 : (gx > NX - 1 ? NX - 1 : gx);
    uint64_t gp = (uint64_t)(uintptr_t)(perm + (size_t)b * NX * NY + (size_t)gx * NY + (lane << 2));
    async_ld_b128(l_perm + (uint32_t)(r * NY * 4) + lrow, gp);
    uint64_t gu = (uint64_t)(uintptr_t)(pressure + ((size_t)(b * T_ + t0) * NX + gx) * NY + (lane << 2));
    async_ld_b128(l_p0 + (uint32_t)(r * NY * 4) + lrow, gu);
  }

  // ---- per-b elementwise inputs (8 elements per thread, rows stride 2) ----
  const int col = tid & 127;
  const int xrb = tid >> 7;              // 0 or 1
  const int ym  = col > 0      ? col - 1 : 0;
  const int yp  = col < NY - 1 ? col + 1 : NY - 1;

  float finv[8], finw[8], dtin[8], phiv[8], wprev[8];
  for (int k = 0; k < 8; ++k) {
    int x = x0 + xrb + 2 * k;
    size_t bi = ((size_t)b * NX + x) * NY + col;
    finv[k] = Q[bi]     * 5000.0f;       // UIR
    finw[k] = Qw[bi]    * 5000.0f;
    dtin[k] = Timev[bi] * 6000.0f;       // MAXZ
    phiv[k] = Phi[bi];
    wprev[k] = (t0 == 0) ? siniuse
             : wsat[((size_t)(b * T_ + t0 - 1) * NX + x) * NY + col];
  }
  async_wait0();
  __syncthreads();

  // ---- pipelined time loop: kick async fetch of t+1, compute t from LDS ----
  for (int tt = 0; tt < TT; ++tt) {
    const int t   = t0 + tt;
    const int cur = tt & 1;

    if (tt + 1 < TT) {
      const uint32_t l_nxt = (cur ^ 1) ? l_p1 : l_p0;
      for (int r = wid; r < HROWS; r += 8) {
        int gx = x0 - 1 + r; gx = gx < 0 ? 0 : (gx > NX - 1 ? NX - 1 : gx);
        uint64_t gu = (uint64_t)(uintptr_t)(pressure + ((size_t)(b * T_ + t + 1) * NX + gx) * NY + (lane << 2));
        async_ld_b128(l_nxt + (uint32_t)(r * NY * 4) + lrow, gu);
      }
    }

    const float* sp = s_p[cur];
#pragma unroll
    for (int k = 0; k < 8; ++k) {
      const int xr = xrb + 2 * k;        // tile row 0..15
      const int lr = xr + 1;             // LDS row (halo offset)

      float pc  = sp[lr * NY + col];
      float pxm = sp[(lr - 1) * NY + col];
      float pxp = sp[(lr + 1) * NY + col];
      float pym = sp[lr * NY + ym];
      float pyp = sp[lr * NY + yp];

      float dudx = (pxp - pxm) * (FD1 * PINIA);
      float dudy = (pyp - pym) * (FD1 * PINIA);
      float d2ux = (pxp - 2.0f * pc + pxm) * (FD2 * PINIA);
      float d2uy = (pyp - 2.0f * pc + pym) * (FD2 * PINIA);

      float dpx = (s_perm[(lr + 1) * NY + col] - s_perm[(lr - 1) * NY + col]) * FD1;
      float dpy = (s_perm[lr * NY + yp]        - s_perm[lr * NY + ym])        * FD1;
      float pv  = s_perm[lr * NY + col];

      size_t idx = ((size_t)(b * T_ + t) * NX + (x0 + xr)) * NY + col;
      float ws  = wsat[idx];
      float wp  = wprev[k];
      float dsw = fmaxf(ws - wp, 0.001f);
      float S   = (wp - 0.1f) * 1.25f;
      float Mw  = S * S;
      float oms = 1.0f - S;
      float Mo  = oms * oms * (1.0f / 2.75f);
      float av  = 500.0f * pv;           // a = m*perm + b, m=500, b=0
      float a1  = (Mw + Mo) * av;
      float a1w = Mw * av;

      float pl   = SCALE * (finv[k] + (c0 * dpx) * dudx + a1 * d2ux + (c0 * dpy) * dudy + a1 * d2uy);
      float flux = (cw0 * dpx) * dudx + a1w * d2ux + (cw0 * dpy) * dudy + a1w * d2uy;
      float sl   = SCALE * (phiv[k] * (dsw / dtin[k]) - (flux + finw[k]));

      // Streaming writes, never re-read: non-temporal so the 126 MB output
      // stream does not evict the reusable ~133 MB input set from the 192 MB L2.
      __builtin_nontemporal_store(pl, &out_p[idx]);
      __builtin_nontemporal_store(sl, &out_s[idx]);
      wprev[k]   = ws;
    }

    async_wait0();      // my wave's async writes into the next buffer have landed
    __syncthreads();    // everyone done reading cur / writing next
  }
}

extern "C" void kernel_launch(void* const* d_in, const int* in_sizes, int n_in,
                              void* d_out, int out_size, void* d_ws, size_t ws_size,
                              hipStream_t stream) {
  // setup_inputs order: pressure, perm, Q, Qw, Time, Pini(unused), Phi, Swini, water_sat
  const float* pressure = (const float*)d_in[0];
  const float* perm     = (const float*)d_in[1];
  const float* Q        = (const float*)d_in[2];
  const float* Qw       = (const float*)d_in[3];
  const float* Timev    = (const float*)d_in[4];
  const float* Phi      = (const float*)d_in[6];
  const float* Swini    = (const float*)d_in[7];
  const float* wsat     = (const float*)d_in[8];
  float* out = (float*)d_out;
  const size_t TOT = (size_t)B_ * T_ * NX * NY;   // p_loss, then s_loss

  dim3 grid(NX / TX, T_ / TT, B_);                // (8, 15, 16) = 1920 blocks
  black_oil_kernel<<<grid, 256, 0, stream>>>(pressure, perm, Q, Qw, Timev,
                                             Phi, Swini, wsat, out, out + TOT);
}